// EventTokenizer_15556371546925
// MI455X (gfx1250) — compile-verified
//
#include <hip/hip_runtime.h>
#include <hip/hip_bf16.h>
#include <math.h>

#define T_FRAMES 262144
#define D_IN 512
#define D_IN4 (D_IN / 4)                  // 128 float4 per row
#define D_MODEL 256
#define TYPE_EMB 32
#define NUM_SCALARS 8
#define K_MAX 64
#define D_Z (D_IN + NUM_SCALARS + TYPE_EMB)   // 552, divisible by 4
#define CONF_TH 0.3f
#define EPS 1e-6f

// workspace layout (floats)
#define WS_FEATSUM 0                          // 64*512
#define WS_SUME    (WS_FEATSUM + K_MAX*D_IN)  // 64
#define WS_PEAK    (WS_SUME + K_MAX)          // 64
#define WS_CONF    (WS_PEAK + K_MAX)          // 64
#define WS_VIS     (WS_CONF + K_MAX)          // 64*3
#define WS_ENT     (WS_VIS + K_MAX*3)         // 64
#define WS_Z       (WS_ENT + K_MAX)           // 64*552

typedef __attribute__((ext_vector_type(2))) float v2f;
typedef __attribute__((ext_vector_type(8))) float v8f;

// ---------------- kernel 1: zero the atomic accumulator region ----------------
__global__ void et_zero_kernel(float* ws) {
    int i = blockIdx.x * blockDim.x + threadIdx.x;
    if (i < K_MAX * D_IN) ws[WS_FEATSUM + i] = 0.0f;
}

// ---------------- kernel 2: feature segment sums (512 MB streaming pass) -----
// grid: (K_MAX, NSLICE); block: 128 threads, each owns one float4 column.
// One block iteration consumes a full 2KB row via global_load_b128 (512B/wave-issue).
#define NSLICE 32
__global__ void et_featsum_kernel(const float4* __restrict__ features,
                                  const int* __restrict__ seg_start,
                                  const int* __restrict__ seg_end,
                                  float* ws) {
    const int s  = blockIdx.x;
    const int c4 = threadIdx.x;                          // 0..127
    const int st = seg_start[s], en = seg_end[s];
    const int len = en - st;
    const int chunk = (len + NSLICE - 1) / NSLICE;
    int t0 = st + blockIdx.y * chunk;
    int t1 = t0 + chunk; if (t1 > en) t1 = en;

    float ax = 0.f, ay = 0.f, az = 0.f, aw = 0.f;
    const float4* p = features + (size_t)t0 * D_IN4 + c4;
    for (int t = t0; t < t1; ++t, p += D_IN4) {
        // 64KB-ahead speculative prefetch (global_prefetch_b8 on gfx1250)
        __builtin_prefetch(p + 32 * D_IN4, 0, 1);
        float4 v = *p;
        ax += v.x; ay += v.y; az += v.z; aw += v.w;
    }
    if (t0 < t1) {
        float* dst = &ws[WS_FEATSUM + s * D_IN + c4 * 4];
        atomicAdd(dst + 0, ax);
        atomicAdd(dst + 1, ay);
        atomicAdd(dst + 2, az);
        atomicAdd(dst + 3, aw);
    }
}

// ---------------- shared-memory block reductions -----------------------------
__device__ __forceinline__ float block_sum(float v, float* sh) {
    int t = threadIdx.x;
    sh[t] = v; __syncthreads();
    for (int o = 128; o > 0; o >>= 1) { if (t < o) sh[t] += sh[t + o]; __syncthreads(); }
    float r = sh[0]; __syncthreads();
    return r;
}
__device__ __forceinline__ float block_max(float v, float* sh) {
    int t = threadIdx.x;
    sh[t] = v; __syncthreads();
    for (int o = 128; o > 0; o >>= 1) { if (t < o) sh[t] = fmaxf(sh[t], sh[t + o]); __syncthreads(); }
    float r = sh[0]; __syncthreads();
    return r;
}

// ---------------- kernel 3: per-segment scalar reductions --------------------
// grid: K_MAX blocks of 256
__global__ void et_scalars_kernel(const float* __restrict__ energy,
                                  const float* __restrict__ frame_conf,
                                  const float* __restrict__ stream_conf,
                                  const int* __restrict__ seg_start,
                                  const int* __restrict__ seg_end,
                                  float* ws) {
    __shared__ float sh[256];
    const int s = blockIdx.x;
    const int st = seg_start[s], en = seg_end[s];
    float se = 0.f, pk = -3.4e38f, cf = 0.f, v0 = 0.f, v1 = 0.f, v2 = 0.f;
    for (int t = st + threadIdx.x; t < en; t += blockDim.x) {
        float e = energy[t];
        se += e;
        pk = fmaxf(pk, e);
        cf += frame_conf[t];
        v0 += stream_conf[t * 3 + 0];
        v1 += stream_conf[t * 3 + 1];
        v2 += stream_conf[t * 3 + 2];
    }
    se = block_sum(se, sh);
    pk = block_max(pk, sh);
    cf = block_sum(cf, sh);
    v0 = block_sum(v0, sh);
    v1 = block_sum(v1, sh);
    v2 = block_sum(v2, sh);
    if (threadIdx.x == 0) {
        ws[WS_SUME + s] = se;
        ws[WS_PEAK + s] = pk;
        ws[WS_CONF + s] = cf;
        ws[WS_VIS + s * 3 + 0] = v0;
        ws[WS_VIS + s * 3 + 1] = v1;
        ws[WS_VIS + s * 3 + 2] = v2;
    }
}

// ---------------- kernel 4: entropy (needs sum_e) ----------------------------
__global__ void et_entropy_kernel(const float* __restrict__ energy,
                                  const int* __restrict__ seg_start,
                                  const int* __restrict__ seg_end,
                                  float* ws) {
    __shared__ float sh[256];
    const int s = blockIdx.x;
    const int st = seg_start[s], en = seg_end[s];
    const float inv = 1.0f / (ws[WS_SUME + s] + EPS);
    float acc = 0.f;
    for (int t = st + threadIdx.x; t < en; t += blockDim.x) {
        float p = energy[t] * inv;
        acc += p * logf(p + EPS);
    }
    acc = block_sum(acc, sh);
    if (threadIdx.x == 0) ws[WS_ENT + s] = -acc;
}

// ---------------- kernel 5: assemble z[64,552] + side outputs ----------------
// grid: K_MAX blocks of 256
__global__ void et_assemble_kernel(const int* __restrict__ seg_start,
                                   const int* __restrict__ seg_end,
                                   const int* __restrict__ event_type_id,
                                   const float* __restrict__ emb,
                                   const int* __restrict__ fps_p,
                                   float* ws, float* out) {
    const int s = blockIdx.x;
    const int tid = threadIdx.x;
    const float len = (float)(seg_end[s] - seg_start[s]);
    const float inv_len = 1.0f / len;
    float* z = ws + WS_Z + (size_t)s * D_Z;

    // mean features
    for (int c = tid; c < D_IN; c += blockDim.x)
        z[c] = ws[WS_FEATSUM + s * D_IN + c] * inv_len;

    // type embedding
    const int ty = event_type_id[s];
    if (tid < TYPE_EMB)
        z[D_IN + NUM_SCALARS + tid] = emb[ty * TYPE_EMB + tid];

    if (tid == 0) {
        const float fps = (float)fps_p[0];
        float sv[NUM_SCALARS];
        sv[0] = len / fps;                               // duration_sec
        sv[1] = (float)seg_start[s] / fps;               // latency_sec
        sv[2] = ws[WS_SUME + s] * inv_len;               // mean_motion
        sv[3] = ws[WS_PEAK + s];                         // peak_motion
        sv[4] = ws[WS_ENT + s];                          // motion_entropy
        sv[5] = ws[WS_VIS + s * 3 + 0] * inv_len;
        sv[6] = ws[WS_VIS + s * 3 + 1] * inv_len;
        sv[7] = ws[WS_VIS + s * 3 + 2] * inv_len;
        float* out_sk = out + K_MAX * D_MODEL + 3 * K_MAX;   // after tokens/mask/type/conf
        for (int j = 0; j < NUM_SCALARS; ++j) {
            float v = log1pf(sv[j]);
            z[D_IN + j] = v;
            out_sk[s * NUM_SCALARS + j] = v;
        }
        const float cmean = ws[WS_CONF + s] * inv_len;
        out[K_MAX * D_MODEL + s]             = (cmean >= CONF_TH) ? 1.0f : 0.0f; // attn_mask
        out[K_MAX * D_MODEL + K_MAX + s]     = (float)ty;                        // event_type_id
        out[K_MAX * D_MODEL + 2 * K_MAX + s] = cmean;                            // conf
    }
}

// ---------------- kernel 6: tokens = z @ W^T + bias via V_WMMA_F32_16X16X4_F32
// 64x256 output = 4x16 tiles of 16x16; one wave32 per tile; K = 552 = 138 steps of 4.
// grid: 8 blocks of 256 (8 waves/block -> 64 waves)
__global__ void et_wmma_gemm_kernel(const float* __restrict__ ws_z,
                                    const float* __restrict__ W,
                                    const float* __restrict__ bias,
                                    float* __restrict__ tokens) {
    const int tid  = threadIdx.x;
    const int wave = blockIdx.x * (blockDim.x / 32) + (tid >> 5);  // 0..63
    const int lane = tid & 31;
    const int tileM = wave >> 4;          // 0..3
    const int tileN = wave & 15;          // 0..15
    const int half  = lane >> 4;          // 0 or 1
    const int l16   = lane & 15;

    const int m = tileM * 16 + l16;       // A row
    const int n = tileN * 16 + l16;       // B/C column

    v8f c;
    const float b0 = bias[n];
#pragma unroll
    for (int i = 0; i < 8; ++i) c[i] = b0;

    const float* zrow = ws_z + (size_t)m * D_Z;
    const float* wrow = W    + (size_t)n * D_Z;

    for (int k0 = 0; k0 < D_Z; k0 += 4) {
        const int kA = k0 + (half << 1);              // reg0 holds K=0 (lanes 0-15) / K=2 (16-31)
        v2f a, b;
        a.x = zrow[kA];     a.y = zrow[kA + 1];       // reg1 holds K=1 / K=3
        b.x = wrow[kA];     b.y = wrow[kA + 1];       // B[k][n] = W[n][k]
        c = __builtin_amdgcn_wmma_f32_16x16x4_f32(
                /*neg_a=*/false, a, /*neg_b=*/false, b,
                /*c_mod=*/(short)0, c, /*reuse_a=*/false, /*reuse_b=*/false);
    }

#pragma unroll
    for (int r = 0; r < 8; ++r) {
        const int row = tileM * 16 + r + (half << 3);  // VGPR r -> M=r (lanes 0-15), M=r+8 (16-31)
        tokens[(size_t)row * D_MODEL + n] = c[r];
    }
}

extern "C" void kernel_launch(void* const* d_in, const int* in_sizes, int n_in,
                              void* d_out, int out_size, void* d_ws, size_t ws_size,
                              hipStream_t stream) {
    const float*  features    = (const float*)d_in[0];
    const float*  energy      = (const float*)d_in[1];
    const float*  frame_conf  = (const float*)d_in[2];
    const float*  stream_conf = (const float*)d_in[3];
    const int*    seg_start   = (const int*)d_in[4];
    const int*    seg_end     = (const int*)d_in[5];
    const int*    etype       = (const int*)d_in[6];
    const float*  emb         = (const float*)d_in[7];
    const float*  W           = (const float*)d_in[8];
    const float*  bias        = (const float*)d_in[9];
    const int*    fps_p       = (const int*)d_in[10];

    float* ws  = (float*)d_ws;
    float* out = (float*)d_out;

    (void)in_sizes; (void)n_in; (void)out_size; (void)ws_size;

    et_zero_kernel<<<(K_MAX * D_IN + 255) / 256, 256, 0, stream>>>(ws);

    dim3 gFeat(K_MAX, NSLICE);
    et_featsum_kernel<<<gFeat, 128, 0, stream>>>((const float4*)features,
                                                 seg_start, seg_end, ws);

    et_scalars_kernel<<<K_MAX, 256, 0, stream>>>(energy, frame_conf, stream_conf,
                                                 seg_start, seg_end, ws);

    et_entropy_kernel<<<K_MAX, 256, 0, stream>>>(energy, seg_start, seg_end, ws);

    et_assemble_kernel<<<K_MAX, 256, 0, stream>>>(seg_start, seg_end, etype, emb,
                                                  fps_p, ws, out);

    et_wmma_gemm_kernel<<<8, 256, 0, stream>>>(ws + WS_Z, W, bias, out);
}